// AttentivePooling_19559281066750
// MI455X (gfx1250) — compile-verified
//
#include <hip/hip_runtime.h>
#include <math.h>

typedef __attribute__((ext_vector_type(2))) float v2f;
typedef __attribute__((ext_vector_type(8))) float v8f;

#define B_   16
#define S_   2048
#define E_   1024
#define H_   8
#define DH_  64
#define INNER_ 512

// ---------------------------------------------------------------------------
// K1: p[h][e] = scale * sum_d q[h*64+d] * Wk[h*64+d][e],  q = Wq @ token
// ---------------------------------------------------------------------------
__global__ void __launch_bounds__(256) k1_prep(const float* __restrict__ tok,
                                               const float* __restrict__ Wq,
                                               const float* __restrict__ Wkv,
                                               float* __restrict__ p) {
    __shared__ float q[INNER_];
    __shared__ float tk[E_];
    int t = threadIdx.x;
    for (int i = t; i < E_; i += 256) tk[i] = tok[i];
    __syncthreads();
    for (int i = t; i < INNER_; i += 256) {
        const float* wr = &Wq[(size_t)i * E_];
        float acc = 0.f;
        for (int e = 0; e < E_; ++e) acc = fmaf(wr[e], tk[e], acc);
        q[i] = acc;
    }
    __syncthreads();
    const float scale = 0.125f; // 1/sqrt(64)
    for (int i = t; i < H_ * E_; i += 256) {
        int h = i >> 10, e = i & (E_ - 1);
        float acc = 0.f;
        for (int d = 0; d < DH_; ++d)
            acc = fmaf(q[h * DH_ + d], Wkv[(size_t)(h * DH_ + d) * E_ + e], acc);
        p[i] = acc * scale;
    }
}

// ---------------------------------------------------------------------------
// K2: sim[b][h][s] = x[b,s,:] . p[h,:]   (fp32 WMMA 16x16x4, K = 1024)
// one wave -> one 16-row tile of x, N = 16 (h padded to 16, cols>=8 zeroed)
// grid: 32768 rows / 16 rows-per-wave / 8 waves-per-block = 256 blocks
// ---------------------------------------------------------------------------
#define LPSTRIDE 1032  // 1024 + 8 pad: breaks 64-bank conflict at stride 1024

__global__ void __launch_bounds__(256) k2_sim(const float* __restrict__ x,
                                              const float* __restrict__ pg,
                                              float* __restrict__ sim) {
    __shared__ float lp[H_ * LPSTRIDE]; // 33 KB
    int t = threadIdx.x;
    for (int i = t; i < H_ * E_; i += 256)
        lp[(i >> 10) * LPSTRIDE + (i & (E_ - 1))] = pg[i];
    __syncthreads();

    int lane = t & 31;
    int wv   = t >> 5;
    int tile = blockIdx.x * 8 + wv;          // 0..2047
    int rowbase = tile * 16;                 // 16 consecutive global rows
    int m  = lane & 15;                      // A row (= B col index n)
    int kb = (lane >> 4) * 2;                // K sub-offset: 0 for lanes<16, 2 else
    float bmask = (m < H_) ? 1.0f : 0.0f;

    const float* xr = &x[(size_t)(rowbase + m) * E_ + kb];
    const float* bp = &lp[(m & 7) * LPSTRIDE + kb];

    v8f c = {};
    for (int k = 0; k < E_; k += 32) {
        __builtin_prefetch(xr + k + 256, 0, 1);   // global_prefetch_b8, stream x
        #pragma unroll
        for (int kk = 0; kk < 32; kk += 4) {
            v2f a = *(const v2f*)(xr + k + kk);   // A[m][k+kb], A[m][k+kb+1]
            v2f b = *(const v2f*)(bp + k + kk);   // B[k+kb][n] = p[n][k+kb]
            b *= bmask;                           // zero padded heads 8..15
            c = __builtin_amdgcn_wmma_f32_16x16x4_f32(
                    false, a, false, b, (short)0, c, false, false);
        }
    }

    // C layout: VGPR r -> M = r (lanes 0-15) / 8+r (lanes 16-31), N = lane&15
    int bidx = rowbase >> 11;        // tiles never straddle a batch (2048%16==0)
    int s0   = rowbase & (S_ - 1);
    int mhi  = (lane >> 4) * 8;
    float* so = &sim[((size_t)bidx * 16 + m) * S_ + s0 + mhi];
    #pragma unroll
    for (int r = 0; r < 8; ++r) so[r] = c[r];
}

// ---------------------------------------------------------------------------
// K3: softmax over S per (b,h); normalization folded in. 128 blocks.
// ---------------------------------------------------------------------------
__global__ void __launch_bounds__(256) k3_softmax(float* __restrict__ sim) {
    int b = blockIdx.x >> 3, h = blockIdx.x & 7;
    float* row = &sim[((size_t)b * 16 + h) * S_];
    __shared__ float sd[256];
    int t = threadIdx.x;
    float v[8];
    float mx = -1e30f;
    #pragma unroll
    for (int i = 0; i < 8; ++i) { v[i] = row[t * 8 + i]; mx = fmaxf(mx, v[i]); }
    sd[t] = mx; __syncthreads();
    for (int o = 128; o > 0; o >>= 1) {
        if (t < o) sd[t] = fmaxf(sd[t], sd[t + o]);
        __syncthreads();
    }
    mx = sd[0]; __syncthreads();
    float s = 0.f;
    #pragma unroll
    for (int i = 0; i < 8; ++i) { v[i] = __expf(v[i] - mx); s += v[i]; }
    sd[t] = s; __syncthreads();
    for (int o = 128; o > 0; o >>= 1) {
        if (t < o) sd[t] += sd[t + o];
        __syncthreads();
    }
    float inv = 1.0f / sd[0];
    #pragma unroll
    for (int i = 0; i < 8; ++i) row[t * 8 + i] = v[i] * inv;
}

// ---------------------------------------------------------------------------
// K4: r[b][h][e] = sum_s attn[b,h,s] * x[b,s,e]   (WMMA, K = 2048)
// x is L2-resident (134 MB < 192 MB L2) -> this pass is cheap on HBM.
// Each wave: 4 e-tiles (64 cols) sharing one A stream. 16 b * 16 grp = 32 blocks.
// ---------------------------------------------------------------------------
__global__ void __launch_bounds__(256) k4_weighted(const float* __restrict__ x,
                                                   const float* __restrict__ attn,
                                                   float* __restrict__ rbuf) {
    int t = threadIdx.x, lane = t & 31, wv = t >> 5;
    int grp = blockIdx.x * 8 + wv;       // 0..255
    int b   = grp >> 4;
    int e0  = (grp & 15) * 64;
    int m   = lane & 15;
    int kb  = (lane >> 4) * 2;

    const float* arow = &attn[((size_t)b * 16 + m) * S_ + kb];
    v8f c0 = {}, c1 = {}, c2 = {}, c3 = {};
    #pragma unroll 2
    for (int k = 0; k < S_; k += 4) {
        v2f a = *(const v2f*)(arow + k);                 // attn rows 8..15 are 0
        const float* xk = &x[((size_t)b * S_ + k + kb) * E_ + e0 + m];
        v2f b0, b1, b2, b3;
        b0.x = xk[0];  b0.y = xk[E_];
        b1.x = xk[16]; b1.y = xk[16 + E_];
        b2.x = xk[32]; b2.y = xk[32 + E_];
        b3.x = xk[48]; b3.y = xk[48 + E_];
        c0 = __builtin_amdgcn_wmma_f32_16x16x4_f32(false, a, false, b0, (short)0, c0, false, false);
        c1 = __builtin_amdgcn_wmma_f32_16x16x4_f32(false, a, false, b1, (short)0, c1, false, false);
        c2 = __builtin_amdgcn_wmma_f32_16x16x4_f32(false, a, false, b2, (short)0, c2, false, false);
        c3 = __builtin_amdgcn_wmma_f32_16x16x4_f32(false, a, false, b3, (short)0, c3, false, false);
    }
    int mhi = (lane >> 4) * 8;
    #pragma unroll
    for (int r = 0; r < 8; ++r) {
        float* ro = &rbuf[((size_t)b * 16 + mhi + r) * E_ + e0 + m];
        ro[0] = c0[r]; ro[16] = c1[r]; ro[32] = c2[r]; ro[48] = c3[r];
    }
}

// ---------------------------------------------------------------------------
// K5: out_inner[b][h*64+d] = r[b][h][:] . Wv[h*64+d][:]   (Wv = Wkv rows 512+)
//     y[b][e] = out_inner[b][:] . Wo[e][:] + bo[e]
// one block per batch; 17 MFLOP total.
// ---------------------------------------------------------------------------
__global__ void __launch_bounds__(256) k5_out(const float* __restrict__ rbuf,
                                              const float* __restrict__ Wkv,
                                              const float* __restrict__ Wo,
                                              const float* __restrict__ bo,
                                              float* __restrict__ out) {
    __shared__ float oi[INNER_];
    int b = blockIdx.x, t = threadIdx.x;
    for (int i = t; i < INNER_; i += 256) {
        int h = i >> 6;
        const float* rr = &rbuf[((size_t)b * 16 + h) * E_];
        const float* wv = &Wkv[(size_t)(INNER_ + i) * E_];
        float acc = 0.f;
        for (int e = 0; e < E_; ++e) acc = fmaf(rr[e], wv[e], acc);
        oi[i] = acc;
    }
    __syncthreads();
    for (int e = t; e < E_; e += 256) {
        const float* wo = &Wo[(size_t)e * INNER_];
        float acc = bo[e];
        for (int i = 0; i < INNER_; ++i) acc = fmaf(oi[i], wo[i], acc);
        out[(size_t)b * E_ + e] = acc;
    }
}

// ---------------------------------------------------------------------------
extern "C" void kernel_launch(void* const* d_in, const int* in_sizes, int n_in,
                              void* d_out, int out_size, void* d_ws, size_t ws_size,
                              hipStream_t stream) {
    const float* x   = (const float*)d_in[0];
    const float* tok = (const float*)d_in[1];
    const float* Wq  = (const float*)d_in[2];
    const float* Wkv = (const float*)d_in[3];
    const float* Wo  = (const float*)d_in[4];
    const float* bo  = (const float*)d_in[5];
    float* out = (float*)d_out;

    float* ws   = (float*)d_ws;
    float* p    = ws;                                  //  8*1024
    float* sim  = p   + H_ * E_;                       // 16*16*2048 (attn in-place)
    float* rbuf = sim + (size_t)B_ * 16 * S_;          // 16*16*1024
    // total ws use: ~3.1 MB

    k1_prep    <<<1,   256, 0, stream>>>(tok, Wq, Wkv, p);
    k2_sim     <<<256, 256, 0, stream>>>(x, p, sim);
    k3_softmax <<<128, 256, 0, stream>>>(sim);
    k4_weighted<<<32,  256, 0, stream>>>(x, sim, rbuf);
    k5_out     <<<16,  256, 0, stream>>>(rbuf, Wkv, Wo, bo, out);
}